// NeighborhoodAttention1D_8349416423829
// MI455X (gfx1250) — compile-verified
//
#include <hip/hip_runtime.h>
#include <math.h>

#if __has_include(<hip/amd_detail/amd_gfx1250_TDM.h>)
#include <hip/amd_detail/amd_gfx1250_TDM.h>
#define TDM_HEADER_PRESENT 1
#endif

typedef float v2f __attribute__((ext_vector_type(2)));
typedef float v4f __attribute__((ext_vector_type(4)));
typedef float v8f __attribute__((ext_vector_type(8)));
typedef int   v4i __attribute__((ext_vector_type(4)));

#define KS 7
#define EDIM 64
#define HN 8
#define QKV_RSTRIDE (HN * EDIM)   // 512 floats between consecutive l for fixed (b,h)
#define LSTRIDE 68                // padded LDS row stride (floats) for Q/K/V tiles
#define PSTRIDE 36                // padded LDS row stride for P tile

// ---------------- async global -> LDS staging (gfx1250 path) ----------------
#if __has_builtin(__builtin_amdgcn_global_load_async_to_lds_b128) && \
    __has_builtin(__builtin_amdgcn_s_wait_asynccnt)
#define USE_ASYNC_LDS 1
#else
#define USE_ASYNC_LDS 0
#endif

typedef __attribute__((address_space(1))) v4i* as1_v4i_ptr;  // global int4*
typedef __attribute__((address_space(3))) v4i* as3_v4i_ptr;  // LDS int4*

__device__ __forceinline__ void stage16(const float* g, float* l) {
#if USE_ASYNC_LDS
    // AS1/AS3 pointers via integer casts; low 32 bits of a generic LDS pointer
    // are the LDS byte offset per the aperture rules.
    as1_v4i_ptr gp = (as1_v4i_ptr)(unsigned long long)g;
    as3_v4i_ptr lp = (as3_v4i_ptr)(unsigned int)(unsigned long long)l;
    __builtin_amdgcn_global_load_async_to_lds_b128(gp, lp, 0, 0);
#else
    *(v4f*)l = *(const v4f*)g;
#endif
}

__device__ __forceinline__ void stage_fence() {
#if USE_ASYNC_LDS
    __builtin_amdgcn_s_wait_asynccnt(0);
#endif
}

// ---------------- TDM probe (never launched; arity/type discovery) ----------
#if __has_builtin(__builtin_amdgcn_tensor_load_to_lds) && \
    __has_builtin(__builtin_amdgcn_s_wait_tensorcnt)
typedef unsigned int u32x4 __attribute__((ext_vector_type(4)));
typedef int i32x4 __attribute__((ext_vector_type(4)));
typedef int i32x8 __attribute__((ext_vector_type(8)));
__global__ void tdm_probe_kernel() {
    u32x4 g0 = {0u, 0u, 0u, 0u};
    i32x8 g1 = {0, 0, 0, 0, 0, 0, 0, 0};
    i32x4 g2 = {0, 0, 0, 0};
    i32x4 g3 = {0, 0, 0, 0};
    i32x8 g4 = {0, 0, 0, 0, 0, 0, 0, 0};
    __builtin_amdgcn_tensor_load_to_lds(g0, g1, g2, g3, g4, 0);
    __builtin_amdgcn_s_wait_tensorcnt(0);
}
#endif

// ---------------- main kernel ----------------
__global__ __launch_bounds__(32)
void natten1d_wmma_kernel(const float* __restrict__ qg,
                          const float* __restrict__ kg,
                          const float* __restrict__ vg,
                          const float* __restrict__ rpb,
                          float* __restrict__ outg,   // (B,L,H,E)
                          float* __restrict__ attng,  // (B,H,L,KS)
                          int B, int L)
{
    __shared__ __align__(16) float sQ[16 * LSTRIDE];
    __shared__ __align__(16) float sK[32 * LSTRIDE];
    __shared__ __align__(16) float sV[32 * LSTRIDE];
    __shared__ __align__(16) float sP[16 * PSTRIDE];
    __shared__ float sR[13];

    const int tid    = (int)threadIdx.x;       // 0..31, one full wave32
    const int ntiles = L >> 4;
    const int bid    = (int)blockIdx.x;
    const int tile   = bid % ntiles;
    const int h      = (bid / ntiles) % HN;
    const int b      = bid / (ntiles * HN);
    const int t      = tile << 4;              // first query position of tile

    const size_t base = ((size_t)b * L) * QKV_RSTRIDE + (size_t)h * EDIM;

    if (tid < 13) sR[tid] = rpb[h * 13 + tid];

    // ---- stage Q (16 rows) and K/V (32 rows: t-3 .. t+28, clamped) into LDS ----
    #pragma unroll
    for (int i = tid; i < 16 * 16; i += 32) {
        const int r = i >> 4, c4 = (i & 15) << 2;
        stage16(qg + base + (size_t)(t + r) * QKV_RSTRIDE + c4, &sQ[r * LSTRIDE + c4]);
    }
    #pragma unroll
    for (int i = tid; i < 32 * 16; i += 32) {
        const int r = i >> 4, c4 = (i & 15) << 2;
        int gr = t - 3 + r;
        gr = gr < 0 ? 0 : (gr >= L ? L - 1 : gr);
        const size_t off = base + (size_t)gr * QKV_RSTRIDE + c4;
        stage16(kg + off, &sK[r * LSTRIDE + c4]);
        stage16(vg + off, &sV[r * LSTRIDE + c4]);
    }
    stage_fence();          // s_wait_asynccnt 0
    __syncthreads();

    const int g  = tid >> 4;    // lane group (0: lanes 0-15, 1: lanes 16-31)
    const int ln = tid & 15;

    // ---- S = Q * K^T over the 32-wide key band, two 16x16 WMMA tiles ----
    v8f acc0 = {0.f,0.f,0.f,0.f,0.f,0.f,0.f,0.f};
    v8f acc1 = {0.f,0.f,0.f,0.f,0.f,0.f,0.f,0.f};
    #pragma unroll
    for (int e4 = 0; e4 < 16; ++e4) {
        const int col = (e4 << 2) + (g << 1);
        v2f a  = *(const v2f*)&sQ[ln * LSTRIDE + col];          // A[M=ln][K=col..col+1]
        v2f b0 = *(const v2f*)&sK[ln * LSTRIDE + col];          // B[K][N=ln], keys 0..15
        v2f b1 = *(const v2f*)&sK[(16 + ln) * LSTRIDE + col];   // keys 16..31
        acc0 = __builtin_amdgcn_wmma_f32_16x16x4_f32(false, a, false, b0, (short)0, acc0, false, false);
        acc1 = __builtin_amdgcn_wmma_f32_16x16x4_f32(false, a, false, b1, (short)0, acc1, false, false);
    }

    // ---- scale + rpb + edge mask, scatter S into LDS P tile ----
    #pragma unroll
    for (int vv = 0; vv < 8; ++vv) {
        const int M = vv + (g << 3);   // query row within tile
        {   // band tile 0: jrel = ln
            const int kk    = ln - M;          // neighbor index 0..6 when valid
            const int jglob = t - 3 + ln;
            const bool ok   = (kk >= 0) && (kk < KS) && (jglob >= 0) && (jglob < L);
            const int  ri   = kk < 0 ? 0 : (kk > 6 ? 6 : kk);
            const float sv  = ok ? (acc0[vv] * 0.125f + sR[3 + ri]) : -INFINITY;
            sP[M * PSTRIDE + ln] = sv;
        }
        {   // band tile 1: jrel = 16 + ln  (kk >= 1 and jglob >= 13 always)
            const int kk    = 16 + ln - M;
            const int jglob = t + 13 + ln;
            const bool ok   = (kk < KS) && (jglob < L);
            const int  ri   = kk > 6 ? 6 : kk;
            const float sv  = ok ? (acc1[vv] * 0.125f + sR[3 + ri]) : -INFINITY;
            sP[M * PSTRIDE + 16 + ln] = sv;
        }
    }
    __syncthreads();

    // ---- per-row softmax over the 7 valid entries; zero elsewhere; emit attn ----
    if (tid < 16) {
        const int r = tid;
        float* row = &sP[r * PSTRIDE];
        float m = -INFINITY;
        #pragma unroll
        for (int kk = 0; kk < KS; ++kk) m = fmaxf(m, row[r + kk]);
        float pv[KS];
        float sum = 0.f;
        #pragma unroll
        for (int kk = 0; kk < KS; ++kk) {
            const float p = __expf(row[r + kk] - m);   // exp(-inf) -> 0 at edges
            pv[kk] = p; sum += p;
        }
        const float inv = 1.0f / sum;
        #pragma unroll
        for (int c = 0; c < 32; ++c) row[c] = 0.f;
        float* aout = attng + (((size_t)b * HN + h) * L + (t + r)) * KS;
        #pragma unroll
        for (int kk = 0; kk < KS; ++kk) {
            const float p = pv[kk] * inv;
            row[r + kk] = p;
            aout[kk] = p;
        }
    }
    __syncthreads();

    // ---- out = P * V : 4 N-tiles of 16 cols, band K-dim 0..23 in chunks of 4 ----
    const size_t obase = ((size_t)b * L + t) * QKV_RSTRIDE + (size_t)h * EDIM;
    #pragma unroll
    for (int nt = 0; nt < 4; ++nt) {
        v8f acc = {0.f,0.f,0.f,0.f,0.f,0.f,0.f,0.f};
        #pragma unroll
        for (int c = 0; c < 6; ++c) {
            const int kr = (c << 2) + (g << 1);
            v2f a = *(const v2f*)&sP[ln * PSTRIDE + kr];        // A[M=ln][K=kr..kr+1]
            v2f bb;
            bb.x = sV[(kr + 0) * LSTRIDE + (nt << 4) + ln];     // B[K=kr  ][N=ln]
            bb.y = sV[(kr + 1) * LSTRIDE + (nt << 4) + ln];     // B[K=kr+1][N=ln]
            acc = __builtin_amdgcn_wmma_f32_16x16x4_f32(false, a, false, bb, (short)0, acc, false, false);
        }
        #pragma unroll
        for (int vv = 0; vv < 8; ++vv) {
            const int M = vv + (g << 3);
            outg[obase + (size_t)M * QKV_RSTRIDE + (nt << 4) + ln] = acc[vv];
        }
    }
}

extern "C" void kernel_launch(void* const* d_in, const int* in_sizes, int n_in,
                              void* d_out, int out_size, void* d_ws, size_t ws_size,
                              hipStream_t stream) {
    (void)n_in; (void)out_size; (void)d_ws; (void)ws_size;
    const float* q   = (const float*)d_in[0];
    const float* k   = (const float*)d_in[1];
    const float* v   = (const float*)d_in[2];
    const float* rpb = (const float*)d_in[3];
    const int L = 8192;
    const int B = in_sizes[0] / (L * HN * EDIM);   // = 4
    float* out  = (float*)d_out;
    float* attn = out + (size_t)B * L * HN * EDIM;
    const int ntiles = L / 16;
    dim3 grid(B * HN * ntiles);   // 16384 blocks, 1 wave32 each
    natten1d_wmma_kernel<<<grid, 32, 0, stream>>>(q, k, v, rpb, out, attn, B, L);
}